// HausdorffLoss_84258668413280
// MI455X (gfx1250) — compile-verified
//
#include <hip/hip_runtime.h>
#include <hip/hip_bf16.h>

// Hausdorff-style loss, exact EDT, for [B,H,W] = [8,256,256] float32.
// Key algebraic simplification: ALPHA==2 and both weighting masks are binary,
// so (sqrt(d2)*m)^2 == d2*m  -> no sqrt needed anywhere.
//
// MI455X reasoning:
//  - inputs are 4 MB (trivial vs 23.3 TB/s HBM and 192 MB L2); the hot loop is
//    the (min,+) horizontal transform: 2*B*H*W*W ~ 268M fadd+fmin VALU ops.
//    (min,+) is not a WMMA-expressible semiring, so it runs on VALU with the
//    g^2 row held in LDS (uniform-index broadcast reads: conflict-free).
//  - the weighted row reductions ARE (+,x) contractions, so wave-level sums
//    use V_WMMA_F32_16X16X4_F32 (f32 exact; distances reach ~1e12 via the BIG
//    sentinel, ruling out f16/bf16): A = 32 lane values (16x4 f32 = 2 VGPRs),
//    B = ones, C = 0  =>  every D column holds the 16 row sums.

#define B_   8
#define H_   256
#define W_   256
#define BH_  (B_ * H_)
#define BW_  (B_ * W_)
#define BHW_ (B_ * H_ * W_)
#define BIGD 1.0e6f

typedef __attribute__((ext_vector_type(2))) float v2f;
typedef __attribute__((ext_vector_type(8))) float v8f;

// ---------------------------------------------------------------------------
// Kernel A: vertical two-pass 1D EDT per column -> g^2 (squared row distance)
// One thread per (which, b, w) column; lanes walk consecutive w => coalesced.
// which==0: feature = pred = (prob > 0);  which==1: feature = (mask > 0).
// ---------------------------------------------------------------------------
__global__ void edt_vertical_kernel(const float* __restrict__ prob,
                                    const float* __restrict__ mask,
                                    float* __restrict__ g2) {
  int t = blockIdx.x * blockDim.x + threadIdx.x;
  if (t >= 2 * BW_) return;
  int which = t / BW_;
  int c = t - which * BW_;
  int b = c / W_;
  int w = c - b * W_;

  const float* __restrict__ src = which ? mask : prob;
  const size_t base = (size_t)b * H_ * W_ + (size_t)w;
  float* __restrict__ out = g2 + (size_t)which * BHW_ + base;

  // forward scan: d = feature ? 0 : d+1  (init BIG, matches reference)
  float d = BIGD;
  for (int i = 0; i < H_; ++i) {
    bool f = src[base + (size_t)i * W_] > 0.0f;
    d = f ? 0.0f : d + 1.0f;
    out[(size_t)i * W_] = d;  // stash d_fwd
  }
  // backward scan, combine, square
  d = BIGD;
  for (int i = H_ - 1; i >= 0; --i) {
    bool f = src[base + (size_t)i * W_] > 0.0f;
    d = f ? 0.0f : d + 1.0f;
    float g = fminf(d, out[(size_t)i * W_]);
    out[(size_t)i * W_] = g * g;
  }
}

// ---------------------------------------------------------------------------
// Kernel B: per row, exact min-plus d2[j] = min_k g2[k] + (j-k)^2, weight,
// then reduce the 256 values to one partial. Wave-level sum uses f32 WMMA.
// blockIdx.x = which*BH + (b*H + i); 256 threads (8 wave32).
// ---------------------------------------------------------------------------
__global__ void minplus_reduce_kernel(const float* __restrict__ prob,
                                      const float* __restrict__ mask,
                                      const float* __restrict__ g2,
                                      float* __restrict__ partials) {
  __shared__ float s_g2[W_];
  __shared__ float s_wave[8];

  int r = blockIdx.x;              // 0 .. 2*BH-1
  int which = r / BH_;             // block-uniform
  int row = r - which * BH_;       // b*H + i
  int j = threadIdx.x;             // column 0..255
  size_t rowbase = (size_t)row * W_;

  s_g2[j] = g2[(size_t)which * BHW_ + rowbase + (size_t)j];
  __syncthreads();

  // (min,+) transform: VALU, LDS broadcast reads (uniform k across the wave)
  float jf = (float)j;
  float m = 3.0e38f;
#pragma unroll 8
  for (int k = 0; k < W_; ++k) {
    float diff = jf - (float)k;
    m = fminf(m, s_g2[k] + diff * diff);
  }

  // binary weight: (d*w)^2 == d2*w for w in {0,1}
  float wgt = (which == 0) ? mask[rowbase + (size_t)j]
                           : ((prob[rowbase + (size_t)j] > 0.0f) ? 1.0f : 0.0f);
  float val = m * wgt;

  // ---- wave32 sum via V_WMMA_F32_16X16X4_F32: D = A x Ones + 0 ----
  // A layout (32-bit 16x4): VGPR0 lane l<16 -> A[l][0], lane l>=16 -> A[l-16][2]
  // so {val, 0} per lane fills 32 distinct A slots; every D column holds the
  // 16 row sums => column total (lane l + lane l^16) = sum of all 32 vals.
  v2f a;  a.x = val;  a.y = 0.0f;
  v2f bm; bm.x = 1.0f; bm.y = 1.0f;
  v8f cz = {};
  v8f dm = __builtin_amdgcn_wmma_f32_16x16x4_f32(
      /*neg_a=*/false, a, /*neg_b=*/false, bm,
      /*c_mod=*/(short)0, cz, /*reuse_a=*/false, /*reuse_b=*/false);

  float s = dm[0] + dm[1] + dm[2] + dm[3] + dm[4] + dm[5] + dm[6] + dm[7];
  s += __shfl_xor(s, 16, 32);      // lanes l and l^16 combine -> wave total

  int lane = j & 31, wave = j >> 5;
  if (lane == 0) s_wave[wave] = s;
  __syncthreads();
  if (j == 0) {
    float tot = 0.0f;
    for (int q = 0; q < 8; ++q) tot += s_wave[q];  // fixed order: deterministic
    partials[r] = tot;
  }
}

// ---------------------------------------------------------------------------
// Kernel C: deterministic finalize. partials[0..BH) -> dist1 (pred-EDT * mask),
// partials[BH..2BH) -> dist2 (gt-EDT * pred). out = 2*dist1 + dist2.
// ---------------------------------------------------------------------------
__global__ void finalize_kernel(const float* __restrict__ partials,
                                float* __restrict__ out) {
  __shared__ float s0[256];
  __shared__ float s1[256];
  int t = threadIdx.x;
  float a0 = 0.0f, a1 = 0.0f;
  for (int k = 0; k < BH_ / 256; ++k) {       // 8 strided, fixed order
    a0 += partials[t + k * 256];
    a1 += partials[BH_ + t + k * 256];
  }
  s0[t] = a0; s1[t] = a1;
  __syncthreads();
  for (int off = 128; off > 0; off >>= 1) {   // fixed-shape tree: deterministic
    if (t < off) { s0[t] += s0[t + off]; s1[t] += s1[t + off]; }
    __syncthreads();
  }
  if (t == 0) {
    const float invN = 1.0f / (float)BHW_;
    float dist1 = s0[0] * invN;
    float dist2 = s1[0] * invN;
    out[0] = 2.0f * dist1 + dist2;            // WEIGHT_FN * dist1 + dist2
  }
}

extern "C" void kernel_launch(void* const* d_in, const int* in_sizes, int n_in,
                              void* d_out, int out_size, void* d_ws, size_t ws_size,
                              hipStream_t stream) {
  const float* prob = (const float*)d_in[0];   // probability_map [8,256,256] f32
  const float* mask = (const float*)d_in[1];   // target_mask    [8,256,256] f32
  float* out = (float*)d_out;                  // scalar f32

  // workspace: g2 for both EDTs (2*BHW floats = 4 MB) + 4096 row partials
  float* g2 = (float*)d_ws;
  float* partials = g2 + (size_t)2 * BHW_;

  // A: 2*B*W = 4096 columns -> 16 blocks x 256
  edt_vertical_kernel<<<(2 * BW_ + 255) / 256, 256, 0, stream>>>(prob, mask, g2);
  // B: one block per (which,row): 4096 blocks x 256
  minplus_reduce_kernel<<<2 * BH_, 256, 0, stream>>>(prob, mask, g2, partials);
  // C: single block finalize
  finalize_kernel<<<1, 256, 0, stream>>>(partials, out);
}